// Loss2D_57432302682561
// MI455X (gfx1250) — compile-verified
//
#include <hip/hip_runtime.h>
#include <math.h>

#define NPTS 68
#define NT   ((NPTS + 15) / 16)   // 5 row tiles of 16 points

typedef __attribute__((ext_vector_type(2))) float v2f;
typedef __attribute__((ext_vector_type(8))) float v8f;

// One wave (32 lanes) handles 8 views.
// B matrix (4x16, f32, per ISA layout): column c = 2*j + e holds Minv_j[:, e]
//   VGPR0 (b.x): lanes 0-15 -> K=0 row, lanes 16-31 -> K=2 row
//   VGPR1 (b.y): lanes 0-15 -> K=1 row, lanes 16-31 -> K=3 row (zero pad)
// A matrix (16x4, f32): lane m (mod 16) holds point row (tile_base + m)
// D (16x16 f32, 8 VGPRs): VGPR r, lane L -> M = r + 8*(L>=16), N = L%16.
__global__ __launch_bounds__(256) void loss2d_view_kernel(
    const float* __restrict__ px,   // 68 x 3
    const float* __restrict__ Y,    // (V+1) x 68 x 15
    float* __restrict__ view_loss,  // V
    int V)
{
    const int lane        = threadIdx.x & 31;
    const int waveInBlock = threadIdx.x >> 5;
    const int waveId      = blockIdx.x * (blockDim.x >> 5) + waveInBlock;
    const int vbase       = waveId * 8;

    // [wave][view j][ Minv(9 row-major) , t(3) ]
    __shared__ float sM[8][8][12];

    // --- lanes 0..7: load per-view params, compute Minv = inv(scale*R) = inv(R)/scale ---
    if (lane < 8) {
        int v  = vbase + lane;
        int vc = (v < V) ? v : (V - 1);
        long prow = (long)(vc + 1) * NPTS * 15;   // y[v, 0, :] global row
        float sc  = Y[prow + 2];
        float r00 = Y[prow + 3],  r01 = Y[prow + 4],  r02 = Y[prow + 5];
        float r10 = Y[prow + 6],  r11 = Y[prow + 7],  r12 = Y[prow + 8];
        float r20 = Y[prow + 9],  r21 = Y[prow + 10], r22 = Y[prow + 11];
        float t0  = Y[prow + 12], t1  = Y[prow + 13], t2  = Y[prow + 14];

        float c00 = r11*r22 - r12*r21;
        float c01 = r12*r20 - r10*r22;
        float c02 = r10*r21 - r11*r20;
        float det = r00*c00 + r01*c01 + r02*c02;
        float inv = 1.0f / (det * sc);

        float* p = sM[waveInBlock][lane];
        p[0] = c00 * inv;                   // Minv[0][0]
        p[1] = (r02*r21 - r01*r22) * inv;   // Minv[0][1]
        p[2] = (r01*r12 - r02*r11) * inv;   // Minv[0][2]
        p[3] = c01 * inv;                   // Minv[1][0]
        p[4] = (r00*r22 - r02*r20) * inv;   // Minv[1][1]
        p[5] = (r02*r10 - r00*r12) * inv;   // Minv[1][2]
        p[6] = c02 * inv;                   // Minv[2][0]
        p[7] = (r01*r20 - r00*r21) * inv;   // Minv[2][1]
        p[8] = (r00*r11 - r01*r10) * inv;   // Minv[2][2]
        p[9] = t0; p[10] = t1; p[11] = t2;
    }
    __syncthreads();

    // --- build B operand + per-column bias t.Minv[:,e] ---
    const int c    = lane & 15;   // N column of B / D
    const int half = lane >> 4;
    const int j    = c >> 1;      // view within group
    const int e    = c & 1;       // output coordinate
    const float* mp = sM[waveInBlock][j];

    v2f b;
    b.x = half ? mp[6 + e] : mp[0 + e];   // K=2 : K=0
    b.y = half ? 0.0f      : mp[3 + e];   // K=3(pad) : K=1
    const float tm = mp[9]*mp[0+e] + mp[10]*mp[3+e] + mp[11]*mp[6+e];

    // --- hoist A operands for all tiles (points_x is tiny & loop invariant) ---
    float a0t[NT], a1t[NT];
#pragma unroll
    for (int tIdx = 0; tIdx < NT; ++tIdx) {
        int arow = tIdx * 16 + (lane & 15);
        if (arow > NPTS - 1) arow = NPTS - 1;          // pads masked later
        a0t[tIdx] = half ? px[arow*3 + 2] : px[arow*3 + 0];
        a1t[tIdx] = half ? 0.0f           : px[arow*3 + 1];
    }

    const int  vj    = vbase + j;
    const int  vload = (vj < V) ? vj : (V - 1);
    const float* yp  = Y + (long)(vload + 1) * NPTS * 15 + e;  // pt_y[v,:,e] base
    const float* yph = yp + half * (8 * 15);   // per-lane base; tiles 0-3 use const offsets

    float sum_d = 0.0f, sum_m = 0.0f;

#pragma unroll
    for (int tIdx = 0; tIdx < NT; ++tIdx) {
        v2f a;
        a.x = a0t[tIdx];
        a.y = a1t[tIdx];

        v8f acc = {};
        acc = __builtin_amdgcn_wmma_f32_16x16x4_f32(
            /*neg_a=*/false, a, /*neg_b=*/false, b,
            /*c_mod=*/(short)0, acc, /*reuse_a=*/false, /*reuse_b=*/false);

        // prefetch next tile's rows (constant offset, always in-bounds for t<3)
        if (tIdx < 3) __builtin_prefetch(yph + (tIdx + 1) * (16 * 15), 0, 1);

#pragma unroll
        for (int r = 0; r < 8; ++r) {
            float pty;
            if (tIdx * 16 + r + 8 <= NPTS - 1) {
                // compile-time: both halves in-bounds -> constant-offset load
                pty = yph[tIdx * (16 * 15) + r * 15];
            } else {
                // last tile: runtime clamp + select (pads get -1 -> mask 0)
                int  row  = tIdx * 16 + r + half * 8;
                int  rowc = (row < NPTS) ? row : (NPTS - 1);
                float l   = yp[rowc * 15];
                pty = (row < NPTS) ? l : -1.0f;
            }

            float xp  = acc[r] - tm;                   // x_proj[row][e] for view j
            float dlt = pty - xp;
            float s   = dlt * dlt;                     // s >= 0
            int  flag = (pty >= 0.0f) ? 1 : 0;
            // carry flag in sign bit (works even for s==0 via -0.0)
            float stag = flag ? -s : s;
            float o    = __shfl_xor(stag, 1, 32);      // exchange with e-partner
            int   ob   = __float_as_int(o);
            float s2   = s + __int_as_float(ob & 0x7fffffff);
            int flagor = flag | (int)((unsigned)ob >> 31);  // (y0>=0)|(y1>=0)
            float f = (float)flagor;
            sum_d += __builtin_amdgcn_sqrtf(s2) * f;
            sum_m += f;
        }
    }

    // merge lane halves (same view j lives in lanes c and c+16)
    sum_d += __shfl_xor(sum_d, 16, 32);
    sum_m += __shfl_xor(sum_m, 16, 32);

    if ((lane & 1) == 0 && lane < 16 && vj < V) {
        view_loss[vj] = sum_d / sum_m;
    }
}

// Deterministic single-block reduction: sum(view_loss) / V^2
__global__ __launch_bounds__(256) void loss2d_reduce_kernel(
    const float* __restrict__ vl, float* __restrict__ out, int V)
{
    __shared__ double sred[256];
    double s = 0.0;
    for (int i = threadIdx.x; i < V; i += 256) s += (double)vl[i];
    sred[threadIdx.x] = s;
    __syncthreads();
    for (int off = 128; off > 0; off >>= 1) {
        if ((int)threadIdx.x < off) sred[threadIdx.x] += sred[threadIdx.x + off];
        __syncthreads();
    }
    if (threadIdx.x == 0) out[0] = (float)(sred[0] / ((double)V * (double)V));
}

extern "C" void kernel_launch(void* const* d_in, const int* in_sizes, int n_in,
                              void* d_out, int out_size, void* d_ws, size_t ws_size,
                              hipStream_t stream) {
    const float* px = (const float*)d_in[0];   // points_x: 68*3
    const float* Y  = (const float*)d_in[1];   // points_y: (V+1)*68*15
    const int V = in_sizes[1] / (NPTS * 15) - 1;

    float* vl = (float*)d_ws;                  // V floats of scratch

    const int waves  = (V + 7) / 8;            // 8 views per wave
    const int blocks = (waves + 7) / 8;        // 8 waves (256 threads) per block
    loss2d_view_kernel<<<blocks, 256, 0, stream>>>(px, Y, vl, V);
    loss2d_reduce_kernel<<<1, 256, 0, stream>>>(vl, (float*)d_out, V);
}